// MDNLoss_34557306864314
// MI455X (gfx1250) — compile-verified
//
#include <hip/hip_runtime.h>
#include <math.h>

// Problem dims (fixed by the reference).
#define BB 512
#define KK 256
#define NN 512
#define CC 6
#define NTILE (NN / 16)            // 32 track tiles of 16 rows per batch
#define WPB 4                      // waves (track tiles) per block
#define LOG2PI 1.8378770664093453f

typedef __attribute__((ext_vector_type(2))) float v2f;
typedef __attribute__((ext_vector_type(8))) float v8f;

// ---------------------------------------------------------------------------
// Kernel 1: per-(b,k) GEMM coefficients + CE table.
//
// logits(n,k) = T(n,:) . W(:,k) with
//   T = [x0^2, x1^2, x0*x1, x0, x1, 1, 0, 0]
//   p = 1/a, r = 1/c, q = -b/(a c), s = q*mu0 + r*mu1
//   W = [-(p^2+q^2)/2, -r^2/2, -q r, p^2 mu0 + q s, r s,
//        log(pi) - log(2pi) - log a - log c - (p^2 mu0^2 + s^2)/2, 0, 0]
// cep(b,k,c) = -log_softmax(ef_logits)[b,k,c]
// ---------------------------------------------------------------------------
__global__ __launch_bounds__(256) void prep_kernel(
    const float* __restrict__ pi, const float* __restrict__ mu,
    const float* __restrict__ L, const float* __restrict__ ef_logits,
    float* __restrict__ W, float* __restrict__ cep) {
  int idx = blockIdx.x * blockDim.x + threadIdx.x;
  if (idx >= BB * KK) return;

  float piv = pi[idx];
  float mu0 = mu[2 * idx + 0], mu1 = mu[2 * idx + 1];
  float a  = L[4 * idx + 0];      // L[...,0,0]
  float bl = L[4 * idx + 2];      // L[...,1,0]
  float cl = L[4 * idx + 3];      // L[...,1,1]

  float p = 1.0f / a, r = 1.0f / cl;
  float q = -bl * p * r;
  float s = q * mu0 + r * mu1;
  float p2 = p * p;

  float* w = W + (size_t)idx * 8;
  w[0] = -0.5f * (p2 + q * q);
  w[1] = -0.5f * r * r;
  w[2] = -q * r;
  w[3] = p2 * mu0 + q * s;
  w[4] = r * s;
  w[5] = __logf(piv) - LOG2PI - __logf(a) - __logf(cl)
         - 0.5f * (p2 * mu0 * mu0 + s * s);
  w[6] = 0.0f;
  w[7] = 0.0f;

  const float* el = ef_logits + (size_t)idx * CC;
  float mx = el[0];
#pragma unroll
  for (int c = 1; c < CC; ++c) mx = fmaxf(mx, el[c]);
  float sum = 0.0f;
#pragma unroll
  for (int c = 0; c < CC; ++c) sum += __expf(el[c] - mx);
  float lse = mx + __logf(sum);
  float* cp = cep + (size_t)idx * CC;
#pragma unroll
  for (int c = 0; c < CC; ++c) cp[c] = lse - el[c];   // = -log_softmax
}

// ---------------------------------------------------------------------------
// Kernel 2: per-batch Lambda = sum_k pi  and  gt_count = sum_n mask.
// ---------------------------------------------------------------------------
__global__ __launch_bounds__(256) void stats_kernel(
    const float* __restrict__ pi, const unsigned char* __restrict__ mask,
    float* __restrict__ lam_out, float* __restrict__ cnt_out) {
  int b = blockIdx.x;
  int t = threadIdx.x;   // 256 threads; K == 256, N == 512
  __shared__ float sl[256];
  __shared__ float sc[256];
  float lam = pi[(size_t)b * KK + t];
  float cnt = (mask[(size_t)b * NN + t] ? 1.0f : 0.0f) +
              (mask[(size_t)b * NN + 256 + t] ? 1.0f : 0.0f);
  sl[t] = lam;
  sc[t] = cnt;
  __syncthreads();
  for (int off = 128; off > 0; off >>= 1) {
    if (t < off) { sl[t] += sl[t + off]; sc[t] += sc[t + off]; }
    __syncthreads();
  }
  if (t == 0) { lam_out[b] = sl[0]; cnt_out[b] = sc[0]; }
}

// ---------------------------------------------------------------------------
// Kernel 3: fused main pass. 4 waves per block; each wave owns one 16-track
// tile of one batch. Pass A: 32 x V_WMMA_F32_16X16X4_F32 build the full
// 16x256 logit strip in VGPRs + max (no exps). Pass B: one exp per logit,
// CE-weighted sums via LDS gathers, add-only butterflies.
// ---------------------------------------------------------------------------
__global__ __launch_bounds__(32 * WPB) void mdn_main_kernel(
    const float* __restrict__ W, const float* __restrict__ cep,
    const float* __restrict__ tracks, const unsigned char* __restrict__ mask,
    float* __restrict__ spat_out, float* __restrict__ ef_out) {
  int tid  = threadIdx.x;
  int wave = tid >> 5;
  int lane = tid & 31;
  int half = lane >> 4;             // 0: C rows 0-7, 1: C rows 8-15
  int col  = lane & 15;             // k column within chunk; also A row
  int b    = blockIdx.y;
  int tile = blockIdx.x * WPB + wave;
  int n0   = tile * 16;

  __shared__ float sCep[KK * CC];   // per-batch -logsoftmax table, 6 KB
  __shared__ float sT[WPB][16][8];  // per-wave 16x8 feature tiles
  __shared__ float sMask[WPB][16];
  __shared__ int   sGt[WPB][16];

  // ---- Stage per-batch CE table into LDS (shared by all 4 waves). ----
  const float* cepB = cep + (size_t)b * (KK * CC);   // 1536 floats, 16B align
#if defined(__HIP_DEVICE_COMPILE__) && \
    __has_builtin(__builtin_amdgcn_global_load_async_to_lds_b128)
  typedef int v4i __attribute__((vector_size(16)));
  typedef __attribute__((address_space(1))) v4i as1_v4i;
  typedef __attribute__((address_space(3))) v4i as3_v4i;
  for (int i = tid * 4; i < KK * CC; i += 32 * WPB * 4) {   // 3 iters
    __builtin_amdgcn_global_load_async_to_lds_b128(
        (as1_v4i*)(cepB + i), (as3_v4i*)&sCep[i], 0, 0);
  }
#if __has_builtin(__builtin_amdgcn_s_wait_asynccnt)
  __builtin_amdgcn_s_wait_asynccnt(0);
#else
  asm volatile("s_wait_asynccnt 0x0" ::: "memory");
#endif
#else
  for (int i = tid; i < (KK * CC) / 4; i += 32 * WPB) {     // 3 iters
    ((float4*)sCep)[i] = ((const float4*)cepB)[i];
  }
#endif

  // ---- Build this wave's 16x8 feature tile. ----
  if (lane < 16) {
    int n = n0 + lane;
    const float* tp = tracks + ((size_t)b * NN + n) * 6;
    float x0 = tp[0], x1 = tp[1];
    sT[wave][lane][0] = x0 * x0;
    sT[wave][lane][1] = x1 * x1;
    sT[wave][lane][2] = x0 * x1;
    sT[wave][lane][3] = x0;
    sT[wave][lane][4] = x1;
    sT[wave][lane][5] = 1.0f;
    sT[wave][lane][6] = 0.0f;
    sT[wave][lane][7] = 0.0f;
    sGt[wave][lane]   = (int)tp[5];
    sMask[wave][lane] = mask[(size_t)b * NN + n] ? 1.0f : 0.0f;
  }
  __syncthreads();

  // A operand (16x4 f32): lane -> row = lane&15; VGPR j holds K = j + 2*half.
  v2f A0 = *(const v2f*)&sT[wave][col][2 * half];       // features 0..3
  v2f A1 = *(const v2f*)&sT[wave][col][4 + 2 * half];   // features 4..7

  int   gt[8];
  float mk[8];
#pragma unroll
  for (int v = 0; v < 8; ++v) {
    gt[v] = sGt[wave][v + 8 * half];     // C-layout row m = v + 8*half
    mk[v] = sMask[wave][v + 8 * half];
  }

  // ---- Pass A: all 256 logits into registers + per-row running max. ----
  v8f   acc[16];
  float mx[8];
#pragma unroll
  for (int v = 0; v < 8; ++v) mx[v] = -INFINITY;

  const float* Wb = W + (size_t)b * (KK * 8);
#pragma unroll
  for (int kc = 0; kc < 16; ++kc) {
    const float* wk = Wb + (size_t)(kc * 16 + col) * 8;
    // B operand (4x16 f32): lanes 0-15 rows {0,1}, lanes 16-31 rows {2,3}.
    v2f B0 = *(const v2f*)(wk + 2 * half);
    v2f B1 = *(const v2f*)(wk + 4 + 2 * half);
    v8f c = {};
    c = __builtin_amdgcn_wmma_f32_16x16x4_f32(false, A0, false, B0,
                                              (short)0, c, false, false);
    acc[kc] = __builtin_amdgcn_wmma_f32_16x16x4_f32(false, A1, false, B1,
                                                    (short)0, c, false, false);
#pragma unroll
    for (int v = 0; v < 8; ++v) mx[v] = fmaxf(mx[v], acc[kc][v]);
  }
  // Row max across the 16 lanes of each half (butterfly stays in-half).
#pragma unroll
  for (int v = 0; v < 8; ++v) {
#pragma unroll
    for (int off = 1; off < 16; off <<= 1)
      mx[v] = fmaxf(mx[v], __shfl_xor(mx[v], off, 32));
  }

  // ---- Pass B: one exp per logit; plain add reductions. ----
  float s[8], E[8];
#pragma unroll
  for (int v = 0; v < 8; ++v) { s[v] = 0.0f; E[v] = 0.0f; }
#pragma unroll
  for (int kc = 0; kc < 16; ++kc) {
    int kk = kc * 16 + col;
#pragma unroll
    for (int v = 0; v < 8; ++v) {
      float e  = __expf(acc[kc][v] - mx[v]);
      float ce = sCep[kk * CC + gt[v]];        // LDS gather, stride-6 banks
      s[v] += e;
      E[v] = fmaf(e, ce, E[v]);
    }
  }
#pragma unroll
  for (int v = 0; v < 8; ++v) {
#pragma unroll
    for (int off = 1; off < 16; off <<= 1) {
      s[v] += __shfl_xor(s[v], off, 32);
      E[v] += __shfl_xor(E[v], off, 32);
    }
  }

  // log_mix = mx + log(s); sum_k resp*ce = E / s. Identical across each half.
  float spat = 0.0f, efp = 0.0f;
#pragma unroll
  for (int v = 0; v < 8; ++v) {
    spat += mk[v] * (mx[v] + __logf(s[v]));
    efp  += mk[v] * (E[v] / s[v]);
  }
  float spat_t = __shfl(spat, 0, 32) + __shfl(spat, 16, 32);
  float ef_t   = __shfl(efp,  0, 32) + __shfl(efp,  16, 32);
  if (lane == 0) {
    int wg = b * NTILE + tile;
    spat_out[wg] = spat_t;
    ef_out[wg]   = ef_t;
  }
}

// ---------------------------------------------------------------------------
// Kernel 4: deterministic final reduction -> 5 outputs.
// ---------------------------------------------------------------------------
__global__ __launch_bounds__(256) void finalize_kernel(
    const float* __restrict__ spat_part, const float* __restrict__ ef_part,
    const float* __restrict__ lam, const float* __restrict__ cnt,
    float* __restrict__ out) {
  int t = threadIdx.x;   // 256
  __shared__ float r0[256], r1[256], r2[256], r3[256], r4[256];
  float sspat = 0.0f, sef = 0.0f, slam = 0.0f, sl1 = 0.0f, snt = 0.0f;
  const int NT = BB * NTILE;
  for (int i = t; i < NT; i += 256) { sspat += spat_part[i]; sef += ef_part[i]; }
  for (int i = t; i < BB; i += 256) {
    float l = lam[i], c = cnt[i];
    slam += l;
    snt  += c;
    sl1  += fabsf(l - c) * sqrtf(c + 1.0f);
  }
  r0[t] = sspat; r1[t] = sef; r2[t] = slam; r3[t] = sl1; r4[t] = snt;
  __syncthreads();
  for (int off = 128; off > 0; off >>= 1) {
    if (t < off) {
      r0[t] += r0[t + off]; r1[t] += r1[t + off]; r2[t] += r2[t + off];
      r3[t] += r3[t + off]; r4[t] += r4[t + off];
    }
    __syncthreads();
  }
  if (t == 0) {
    float n_total    = fmaxf(r4[0], 1.0f);
    float spatial    = -r0[0] / n_total;
    float ef_loss    =  r1[0] / n_total;
    float count_loss =  r2[0] / (float)BB;
    float count_l1   =  r3[0] / (float)BB;
    out[0] = count_loss + spatial + ef_loss + count_l1;
    out[1] = spatial;
    out[2] = count_loss;
    out[3] = count_l1;
    out[4] = ef_loss;
  }
}

// ---------------------------------------------------------------------------
// Launch. Inputs (setup_inputs order): pi, mu, L, ef_logits, tracks,
// track_mask (JAX bool -> 1 byte each). Output: 5 float32.
// ---------------------------------------------------------------------------
extern "C" void kernel_launch(void* const* d_in, const int* in_sizes, int n_in,
                              void* d_out, int out_size, void* d_ws,
                              size_t ws_size, hipStream_t stream) {
  const float*         pi     = (const float*)d_in[0];
  const float*         mu     = (const float*)d_in[1];
  const float*         L      = (const float*)d_in[2];
  const float*         efl    = (const float*)d_in[3];
  const float*         tracks = (const float*)d_in[4];
  const unsigned char* tmask  = (const unsigned char*)d_in[5];

  float* ws     = (float*)d_ws;
  float* W      = ws;                                 // BB*KK*8   = 4 MB
  float* cep    = W + (size_t)BB * KK * 8;            // BB*KK*6   = 3 MB
  float* spat_p = cep + (size_t)BB * KK * CC;         // BB*NTILE
  float* ef_p   = spat_p + (size_t)BB * NTILE;        // BB*NTILE
  float* lam    = ef_p + (size_t)BB * NTILE;          // BB
  float* cnt    = lam + BB;                           // BB

  prep_kernel<<<(BB * KK + 255) / 256, 256, 0, stream>>>(pi, mu, L, efl, W, cep);
  stats_kernel<<<BB, 256, 0, stream>>>(pi, tmask, lam, cnt);
  mdn_main_kernel<<<dim3(NTILE / WPB, BB), 32 * WPB, 0, stream>>>(
      W, cep, tracks, tmask, spat_p, ef_p);
  finalize_kernel<<<1, 256, 0, stream>>>(spat_p, ef_p, lam, cnt, (float*)d_out);
}